// WindLossLayer_35158602285638
// MI455X (gfx1250) — compile-verified
//
#include <hip/hip_runtime.h>

typedef float v2f __attribute__((ext_vector_type(2)));
typedef float v8f __attribute__((ext_vector_type(8)));

#define NSTA 73
#define W 512
#define PI_F 3.14159265358979323846f

__constant__ int c_lat[NSTA] = {
    9, 61, 66, 70, 70, 74, 74, 76, 77, 81, 81, 82, 87, 89, 89, 102, 112, 118,
    118, 125, 144, 160, 160, 169, 185, 185, 189, 189, 189, 205, 206, 210, 211,
    220, 223, 227, 229, 230, 236, 239, 252, 270, 277, 283, 288, 288, 294, 300,
    304, 309, 309, 311, 312, 313, 315, 315, 316, 317, 317, 318, 319, 320, 320,
    320, 320, 320, 321, 321, 324, 326, 330, 331, 332};
__constant__ int c_lon[NSTA] = {
    79, 42, 57, 34, 43, 31, 32, 36, 29, 30, 49, 34, 33, 31, 116, 54, 21, 22,
    117, 30, 32, 25, 45, 35, 21, 55, 26, 35, 68, 69, 41, 97, 160, 55, 68, 47,
    65, 62, 57, 75, 76, 83, 180, 90, 109, 175, 98, 122, 104, 121, 123, 154,
    146, 131, 146, 152, 153, 109, 144, 151, 158, 121, 149, 152, 153, 165, 137,
    150, 152, 177, 145, 169, 153};

__global__ __launch_bounds__(128) void wind_loss_kernel(
    const float* __restrict__ y_pred,
    const float* __restrict__ wind_dir,
    const float* __restrict__ wind_spd,
    float* __restrict__ out) {
  __shared__ float lds[128];

  const int s = threadIdx.x;
  float penalty = 0.0f;

  if (s < NSTA) {
    const int lat = c_lat[s];
    const int lon = c_lon[s];

    // 3x3 patch gather
    float p[3][3];
#pragma unroll
    for (int r = 0; r < 3; ++r)
#pragma unroll
      for (int c = 0; c < 3; ++c)
        p[r][c] = y_pred[(lat - 1 + r) * W + (lon - 1 + c)];

    const float center = p[1][1];

    // exp(|diff|) and signed notation, center exp(0)=1 cancels the -1.0
    float sum9 = 0.0f;
    float sgn[3][3];  // exp(|diff|) * notation (un-normalized)
#pragma unroll
    for (int r = 0; r < 3; ++r)
#pragma unroll
      for (int c = 0; c < 3; ++c) {
        const float diff = p[r][c] - center;
        const float ea = expf(fabsf(diff));
        sum9 += ea;
        // notation = round(diff/|diff+1e-5| * 100)/100, RNE like jnp.round
        const float nota = rintf(diff / fabsf(diff + 1e-5f) * 100.0f) * 0.01f;
        sgn[r][c] = ea * nota;
      }
    const float inv = 1.0f / (sum9 - 1.0f);

    const float theta = wind_dir[lat * W + lon] * (PI_F / 180.0f);
    const float ct = cosf(theta);
    const float st = sinf(theta);

    // d_sum = ct * sum(n_j cos k_j) + st * sum(n_j sin k_j)
    const int XL[8] = {1, 0, 0, 0, 1, 2, 2, 2};
    const int YL[8] = {2, 2, 1, 0, 0, 0, 1, 2};
    float sc = 0.0f, ss = 0.0f;
#pragma unroll
    for (int j = 0; j < 8; ++j) {
      const float kj = (float)j * (PI_F / 4.0f);
      const float nj = sgn[XL[j]][YL[j]] * inv;
      sc = fmaf(nj, cosf(kj), sc);
      ss = fmaf(nj, sinf(kj), ss);
    }
    const float d_sum = ct * sc + st * ss;
    const float similarity = (d_sum + 1.0f) * 0.5f;
    penalty = wind_spd[lat * W + lon] * (1.0f - similarity);
  }

  lds[threadIdx.x] = penalty;  // lanes >= 73 contribute 0
  __syncthreads();

  // ---- 128 -> 1 reduction via two chained V_WMMA_F32_16X16X4_F32 ----
  // A (16x4 f32) layout: lanes 0-15 -> M=lane, K=0,1 in VGPR0/1;
  //                      lanes 16-31 -> M=lane-16, K=2,3.
  // Map A[m][k] = v[4m + k]; B = all ones => D[m][n] = rowsum(m).
  const int l = threadIdx.x & 31;
  const int m = l & 15;
  const int kb = (l >> 4) << 1;

  v2f a0, a1;
  a0.x = lds[4 * m + kb];
  a0.y = lds[4 * m + kb + 1];
  a1.x = lds[64 + 4 * m + kb];
  a1.y = lds[64 + 4 * m + kb + 1];
  v2f b;
  b.x = 1.0f;
  b.y = 1.0f;

  v8f acc = {};
  acc = __builtin_amdgcn_wmma_f32_16x16x4_f32(false, a0, false, b, (short)0,
                                              acc, false, false);
  acc = __builtin_amdgcn_wmma_f32_16x16x4_f32(false, a1, false, b, (short)0,
                                              acc, false, false);

  // Per-lane: sum of 8 D VGPRs = sum of rowsums for this lane's half.
  float part = acc[0] + acc[1] + acc[2] + acc[3] + acc[4] + acc[5] + acc[6] +
               acc[7];
  // Column 0 lives in lane 0 (M=0..7) and lane 16 (M=8..15).
  const float t0 =
      __int_as_float(__builtin_amdgcn_readlane(__float_as_int(part), 0));
  const float t1 =
      __int_as_float(__builtin_amdgcn_readlane(__float_as_int(part), 16));
  const float total = t0 + t1;

  if (threadIdx.x == 0) {
    out[0] = 0.2f * total / 73.0f;
  }
}

extern "C" void kernel_launch(void* const* d_in, const int* in_sizes, int n_in,
                              void* d_out, int out_size, void* d_ws,
                              size_t ws_size, hipStream_t stream) {
  const float* y_pred = (const float*)d_in[0];
  const float* wind_dir = (const float*)d_in[1];
  const float* wind_spd = (const float*)d_in[2];
  float* out = (float*)d_out;
  wind_loss_kernel<<<dim3(1), dim3(128), 0, stream>>>(y_pred, wind_dir,
                                                      wind_spd, out);
}